// diceloss_81956565942483
// MI455X (gfx1250) — compile-verified
//
#include <hip/hip_runtime.h>

// -------- types --------
typedef __attribute__((ext_vector_type(4))) float v4f;
typedef __attribute__((ext_vector_type(2))) float v2f;
typedef __attribute__((ext_vector_type(8))) float v8f;

#define THREADS 256          // 8 wave32 per block
#define NWAVES  (THREADS / 32)
#define MAXBLOCKS 2048

// ---------------------------------------------------------------------------
// Pass 1: streaming masked reduction. Bandwidth-bound: float4 nontemporal
// loads, 3 FMAs + 1 cmp + 2 selects per element. Per-block partial triple
// (intersection, A_sum, B_sum) written to d_ws — deterministic (no atomics).
// ---------------------------------------------------------------------------
__global__ __launch_bounds__(THREADS) void dice_partial(
    const float* __restrict__ pred,
    const float* __restrict__ targ,
    float* __restrict__ part,          // [gridDim.x * 3]
    long long n4,                      // number of float4 elements
    long long n)                       // total scalar elements
{
    const v4f* __restrict__ p4 = (const v4f*)pred;
    const v4f* __restrict__ t4 = (const v4f*)targ;

    float si = 0.f, sa = 0.f, sb = 0.f;

    long long gtid    = (long long)blockIdx.x * blockDim.x + threadIdx.x;
    long long gstride = (long long)gridDim.x * blockDim.x;

    for (long long i = gtid; i < n4; i += gstride) {
        v4f p = __builtin_nontemporal_load(&p4[i]);   // global_load_b128, NT
        v4f t = __builtin_nontemporal_load(&t4[i]);
#pragma unroll
        for (int c = 0; c < 4; ++c) {
            float tc = t[c];
            float pc = p[c];
            bool  m  = tc > -1.0f;
            float vi = m ? pc : 0.0f;
            float vt = m ? tc : 0.0f;
            si = fmaf(vi, vt, si);
            sa = fmaf(vi, vi, sa);
            sb = fmaf(vt, vt, sb);
        }
    }
    // scalar tail (n not divisible by 4 — not hit for this shape, but correct)
    for (long long i = 4 * n4 + gtid; i < n; i += gstride) {
        float tc = targ[i];
        float pc = pred[i];
        bool  m  = tc > -1.0f;
        float vi = m ? pc : 0.0f;
        float vt = m ? tc : 0.0f;
        si = fmaf(vi, vt, si);
        sa = fmaf(vi, vi, sa);
        sb = fmaf(vt, vt, sb);
    }

    // wave32 XOR-fold reduction (warpSize == 32 on gfx1250)
#pragma unroll
    for (int off = 16; off > 0; off >>= 1) {
        si += __shfl_xor(si, off, 32);
        sa += __shfl_xor(sa, off, 32);
        sb += __shfl_xor(sb, off, 32);
    }

    __shared__ float s_si[NWAVES], s_sa[NWAVES], s_sb[NWAVES];
    int wave = threadIdx.x >> 5;
    int lane = threadIdx.x & 31;
    if (lane == 0) { s_si[wave] = si; s_sa[wave] = sa; s_sb[wave] = sb; }
    __syncthreads();

    if (threadIdx.x == 0) {
        float a = 0.f, b = 0.f, c = 0.f;
#pragma unroll
        for (int w = 0; w < NWAVES; ++w) { a += s_si[w]; b += s_sa[w]; c += s_sb[w]; }
        part[3 * blockIdx.x + 0] = a;
        part[3 * blockIdx.x + 1] = b;
        part[3 * blockIdx.x + 2] = c;
    }
}

// ---------------------------------------------------------------------------
// Pass 2: single wave32. Each lane accumulates a strided share of the block
// partials, then V_WMMA_F32_16X16X4_F32 performs the 32-lane reduction:
//   A[L%16][2*(L/16)] = x(L), all other A slots 0, B = ones
//   => D[m][n] = x(m) + x(m+16)  (independent of n)
// C/D layout: lane 0 holds D[0..7][0] in d[0..7], lane 16 holds D[8..15][0].
// Sum 8 regs + shfl from lane 16 => exact fp32 total.
// ---------------------------------------------------------------------------
__global__ __launch_bounds__(32) void dice_final(
    const float* __restrict__ part, int nblocks, float* __restrict__ out)
{
    int lane = threadIdx.x;

    float acc0 = 0.f, acc1 = 0.f, acc2 = 0.f;
    for (int i = lane; i < nblocks; i += 32) {
        acc0 += part[3 * i + 0];
        acc1 += part[3 * i + 1];
        acc2 += part[3 * i + 2];
    }

    const v2f Bones = {1.0f, 1.0f};   // every B[k][n] = 1 regardless of layout
    float res[3];
    float accs[3] = {acc0, acc1, acc2};
#pragma unroll
    for (int c = 0; c < 3; ++c) {
        v2f A = {accs[c], 0.0f};      // lane L: A[L%16][2*(L/16)] = x(L)
        v8f C = {};
        // 8 args: (neg_a, A, neg_b, B, c_mod, C, reuse_a, reuse_b)
        v8f D = __builtin_amdgcn_wmma_f32_16x16x4_f32(
            false, A, false, Bones, (short)0, C, false, false);
        float s = D[0] + D[1] + D[2] + D[3] + D[4] + D[5] + D[6] + D[7];
        s += __shfl(s, 16, 32);       // lane 0 <- lane 16's half
        res[c] = s;
    }

    if (lane == 0) {
        float I  = res[0];
        float As = res[1];
        float Bs = res[2];
        out[0] = 1.0f - (2.0f * I + 1.0f) / (As + Bs + 1.0f);
    }
}

// ---------------------------------------------------------------------------
extern "C" void kernel_launch(void* const* d_in, const int* in_sizes, int n_in,
                              void* d_out, int out_size, void* d_ws, size_t ws_size,
                              hipStream_t stream) {
    const float* pred = (const float*)d_in[0];
    const float* targ = (const float*)d_in[1];
    float* out  = (float*)d_out;
    float* part = (float*)d_ws;

    long long n  = (long long)in_sizes[0];
    long long n4 = n / 4;

    int blocks = MAXBLOCKS;
    long long maxb_ws = (long long)(ws_size / (3 * sizeof(float)));
    if (maxb_ws < blocks) blocks = (int)maxb_ws;
    long long needed = (n4 + THREADS - 1) / THREADS;
    if (needed < blocks) blocks = (int)needed;
    if (blocks < 1) blocks = 1;

    dice_partial<<<blocks, THREADS, 0, stream>>>(pred, targ, part, n4, n);
    dice_final<<<1, 32, 0, stream>>>(part, blocks, out);
}